// Aggregation_83614423319320
// MI455X (gfx1250) — compile-verified
//
#include <hip/hip_runtime.h>
#include <cstdint>

// Problem constants (from reference): N=16, C_X=512, H=W=64, C_W=64, K=3, groups=8
#define HWD     64
#define PLANE   4096          // 64*64 elements per (n, channel) plane
#define GROUPS  8
#define CWN     64
#define THREADS 512

// One block per (n, cw). Stages 8 input planes (one per group, same cw) into LDS
// with gfx1250 async global->LDS copies (ASYNCcnt), then computes the 9-tap
// position-dependent weighted sum. All 3x3 boundary handling is folded into the
// 9 weight vectors (amortized 8x across groups) so the inner loop does pure
// wide LDS loads + FMAs.
__global__ __launch_bounds__(THREADS) void agg_kernel(
    const float* __restrict__ input,    // (16, 512, 64, 64)
    const float* __restrict__ weight,   // (16, 64, 9, 4096)
    float* __restrict__ out)            // (16, 512, 64, 64)
{
    extern __shared__ float tile[];     // GROUPS * PLANE floats = 128 KB

    const int tid = threadIdx.x;
    const int n   = blockIdx.x >> 6;    // / 64
    const int cw  = blockIdx.x & 63;

    // ---------------- Stage: async global -> LDS (ASYNCcnt path) ----------------
    const float* in_base = input + ((size_t)n * 512 + (size_t)cw) * PLANE;
    // Low 32 bits of a flat shared-aperture address == LDS byte offset (ISA 10.2).
    const uint32_t lds_base = (uint32_t)(uintptr_t)tile;

    // 8 planes * 1024 float4 = 8192 b128 transfers; 512 threads -> 16 each.
    #pragma unroll
    for (int it = 0; it < (GROUPS * (PLANE / 4)) / THREADS; ++it) {
        const int idx = it * THREADS + tid;      // float4 index across all planes
        const int g   = idx >> 10;               // / 1024
        const int i4  = idx & 1023;
        const uint64_t ga = (uint64_t)(uintptr_t)(in_base + (size_t)g * CWN * PLANE)
                          + (uint64_t)((uint32_t)i4 * 16u);
        const uint32_t la = lds_base + (uint32_t)idx * 16u;
        asm volatile("global_load_async_to_lds_b128 %0, %1, off"
                     :: "v"(la), "v"(ga)
                     : "memory");
    }
    asm volatile("s_wait_asynccnt 0x0" ::: "memory");
    __syncthreads();

    // ---------------- Compute ----------------
    const float* wpt = weight + ((size_t)n * CWN + (size_t)cw) * 9 * PLANE;
    float* out_base  = out + ((size_t)n * 512 + (size_t)cw) * PLANE;

    #pragma unroll
    for (int it = 0; it < (PLANE / 4) / THREADS; ++it) {   // 2 iterations
        const int p4 = it * THREADS + tid;   // float4 pixel index, 0..1023
        const int p  = p4 * 4;
        const int h  = p >> 6;
        const int w0 = p & 63;               // multiple of 4

        // 9 weight vectors for these 4 pixels; reused for all 8 groups.
        float4 wv[9];
        #pragma unroll
        for (int k = 0; k < 9; ++k)
            wv[k] = *(const float4*)(wpt + (size_t)k * PLANE + p);

        // ---- Fold ALL boundary handling into the weights (once per 8 groups) ----
        const float rmT = (h > 0)        ? 1.f : 0.f;   // dh=-1 row valid
        const float rmB = (h < HWD - 1)  ? 1.f : 0.f;   // dh=+1 row valid
        #pragma unroll
        for (int k = 0; k < 3; ++k) {
            wv[k].x *= rmT; wv[k].y *= rmT; wv[k].z *= rmT; wv[k].w *= rmT;
        }
        #pragma unroll
        for (int k = 6; k < 9; ++k) {
            wv[k].x *= rmB; wv[k].y *= rmB; wv[k].z *= rmB; wv[k].w *= rmB;
        }
        const float lmL = (w0 > 0)        ? 1.f : 0.f;  // left neighbor valid
        const float lmR = (w0 < HWD - 4)  ? 1.f : 0.f;  // right neighbor valid
        wv[0].x *= lmL; wv[3].x *= lmL; wv[6].x *= lmL; // j=0, dw=-1 taps
        wv[2].w *= lmR; wv[5].w *= lmR; wv[8].w *= lmR; // j=3, dw=+1 taps

        #pragma unroll
        for (int g = 0; g < GROUPS; ++g) {
            // center address of the 4-pixel segment in this group's plane
            const float* tg = tile + (size_t)g * PLANE + h * HWD + w0;
            float acc0 = 0.f, acc1 = 0.f, acc2 = 0.f, acc3 = 0.f;

            #pragma unroll
            for (int r = 0; r < 3; ++r) {            // dh = r-1
                const float* row = tg + (r - 1) * HWD;
                // Unconditional wide loads; invalid taps have zeroed weights.
                const float4 m  = *(const float4*)row;   // ds_load_b128
                const float  xl = row[-1];               // ds_load_b32
                const float  xr = row[4];                // ds_load_b32
                const int k0 = r * 3;        // dw=-1
                const int k1 = r * 3 + 1;    // dw= 0
                const int k2 = r * 3 + 2;    // dw=+1
                acc0 += xl  * wv[k0].x + m.x * wv[k1].x + m.y * wv[k2].x;
                acc1 += m.x * wv[k0].y + m.y * wv[k1].y + m.z * wv[k2].y;
                acc2 += m.y * wv[k0].z + m.z * wv[k1].z + m.w * wv[k2].z;
                acc3 += m.z * wv[k0].w + m.w * wv[k1].w + xr  * wv[k2].w;
            }
            *(float4*)(out_base + (size_t)g * CWN * PLANE + p) =
                make_float4(acc0, acc1, acc2, acc3);
        }
    }
}

extern "C" void kernel_launch(void* const* d_in, const int* in_sizes, int n_in,
                              void* d_out, int out_size, void* d_ws, size_t ws_size,
                              hipStream_t stream) {
    const float* input  = (const float*)d_in[0];   // (16,512,64,64) fp32
    const float* weight = (const float*)d_in[1];   // (16,64,9,4096) fp32
    float* out = (float*)d_out;                    // (16,512,64,64) fp32

    const size_t lds_bytes = (size_t)GROUPS * PLANE * sizeof(float);  // 128 KB
    // Allow >64KB dynamic LDS (gfx1250 WGP has 320 KB). Host-side attribute set,
    // not a stream op -> safe under graph capture; deterministic every call.
    (void)hipFuncSetAttribute((const void*)agg_kernel,
                              hipFuncAttributeMaxDynamicSharedMemorySize,
                              (int)lds_bytes);

    dim3 grid(16 * CWN);     // 1024 blocks: one per (n, cw)
    dim3 block(THREADS);     // 512 threads = 16 waves (wave32)
    agg_kernel<<<grid, block, lds_bytes, stream>>>(input, weight, out);
}